// GCN_54030688584002
// MI455X (gfx1250) — compile-verified
//
#include <hip/hip_runtime.h>
#include <hip/hip_bf16.h>

typedef float v2f __attribute__((ext_vector_type(2)));
typedef float v8f __attribute__((ext_vector_type(8)));

// ---------------------------------------------------------------------------
// Utility: zero a float buffer
// ---------------------------------------------------------------------------
__global__ void zero_f32(float* __restrict__ p, long long cnt) {
    long long i = (long long)blockIdx.x * blockDim.x + threadIdx.x;
    if (i < cnt) p[i] = 0.0f;
}

// ---------------------------------------------------------------------------
// Degree accumulation (float atomics) + rsqrt normalization
// ---------------------------------------------------------------------------
__global__ void degree_kernel(const int* __restrict__ src, const int* __restrict__ dst,
                              float* __restrict__ out_deg, float* __restrict__ in_deg, int E) {
    int e = blockIdx.x * blockDim.x + threadIdx.x;
    if (e >= E) return;
    atomicAdd(&out_deg[src[e]], 1.0f);
    atomicAdd(&in_deg[dst[e]], 1.0f);
}

__global__ void norm_kernel(float* __restrict__ od, float* __restrict__ id, int n) {
    int i = blockIdx.x * blockDim.x + threadIdx.x;
    if (i >= n) return;
    float a = od[i];
    od[i] = (a > 0.0f) ? rsqrtf(a) : 0.0f;
    float b = id[i];
    id[i] = (b > 0.0f) ? rsqrtf(b) : 0.0f;
}

// ---------------------------------------------------------------------------
// WMMA GEMM with optional fused previous-layer epilogue:
//   FUSE_EPI=0:  Xs[r,c] = X[r,c] * out_norm[r]                  (layer 1)
//   FUSE_EPI=1:  Xs[r,c] = relu(X[r,c]*in_norm[r] + bias[c]) * out_norm[r]
//                where X = scatter accumulator of the previous layer
//   Y = Xs @ W[DIN x 64]
// Block: 256 threads (8 waves), tile 32 rows x 64 cols; each wave owns a
// 16x16 tile, K-loop in steps of 4 via V_WMMA_F32_16X16X4_F32.
// ---------------------------------------------------------------------------
template <int DIN, bool FUSE_EPI>
__global__ __launch_bounds__(256) void gemm_wmma(const float* __restrict__ X,
                                                 const float* __restrict__ W,
                                                 const float* __restrict__ onorm,
                                                 const float* __restrict__ inorm,
                                                 const float* __restrict__ bias,
                                                 float* __restrict__ Y, int n) {
    constexpr int DOUT = 64;
    constexpr int XS = DIN + 1;              // pad to avoid LDS bank conflicts
    __shared__ float sW[DIN * DOUT];
    __shared__ float sX[32 * XS];

    const int tid = threadIdx.x;
    const int row0 = blockIdx.x * 32;

    // Stage W (coalesced)
    for (int i = tid; i < DIN * DOUT; i += 256) sW[i] = W[i];
    // Stage 32 rows of X with fused scaling / previous-layer epilogue
    for (int i = tid; i < 32 * DIN; i += 256) {
        int r = i / DIN;
        int c = i - r * DIN;
        int row = row0 + r;
        float v = 0.0f;
        if (row < n) {
            v = X[(size_t)row * DIN + c];
            if (FUSE_EPI) v = fmaxf(v * inorm[row] + bias[c], 0.0f);
            v *= onorm[row];
        }
        sX[r * XS + c] = v;
    }
    __syncthreads();

    const int wave = tid >> 5;
    const int lane = tid & 31;
    const int half = lane >> 4;   // 0: lanes 0-15, 1: lanes 16-31
    const int l    = lane & 15;
    const int rbase = (wave >> 2) * 16;
    const int cbase = (wave & 3) * 16;

    v8f acc = {};
    #pragma unroll 4
    for (int k0 = 0; k0 < DIN; k0 += 4) {
        const int kk = k0 + 2 * half;
        // A fragment (16x4): lanes 0-15 hold K=k0,k0+1; lanes 16-31 hold K=k0+2,k0+3
        v2f a, b;
        a[0] = sX[(rbase + l) * XS + kk];
        a[1] = sX[(rbase + l) * XS + kk + 1];
        // B fragment (4x16): same half-wave K split; N striped across lanes
        b[0] = sW[kk * DOUT + cbase + l];
        b[1] = sW[(kk + 1) * DOUT + cbase + l];
        acc = __builtin_amdgcn_wmma_f32_16x16x4_f32(
            /*neg_a=*/false, a, /*neg_b=*/false, b,
            /*c_mod=*/(short)0, acc, /*reuse_a=*/false, /*reuse_b=*/false);
    }

    // D layout: VGPR i -> M=i (lanes 0-15), M=i+8 (lanes 16-31); N = lane%16
    const int col = cbase + l;
    const int rowb = row0 + rbase + (half ? 8 : 0);
    if (row0 + 32 <= n) {
        // Fast path: whole tile in range -> branch-free, full-EXEC stores
        #pragma unroll
        for (int i = 0; i < 8; ++i)
            Y[(size_t)(rowb + i) * DOUT + col] = acc[i];
    } else {
        #pragma unroll
        for (int i = 0; i < 8; ++i) {
            int row = rowb + i;
            if (row < n) Y[(size_t)row * DOUT + col] = acc[i];
        }
    }
}

// ---------------------------------------------------------------------------
// Small GEMM for final layer (64 -> 2) with fused layer-4 epilogue:
//   Y[r,c] = out_norm[r] * sum_k relu(agg[r,k]*in_norm[r] + b4[k]) * W[k,c]
// ---------------------------------------------------------------------------
__global__ void gemm_small(const float* __restrict__ agg, const float* __restrict__ W,
                           const float* __restrict__ onorm, const float* __restrict__ inorm,
                           const float* __restrict__ bias, float* __restrict__ Y, int n) {
    int idx = blockIdx.x * blockDim.x + threadIdx.x;
    if (idx >= n * 2) return;
    int row = idx >> 1;
    int c = idx & 1;
    const float* x = agg + (size_t)row * 64;
    const float in = inorm[row];
    float s = 0.0f;
    #pragma unroll
    for (int k = 0; k < 64; ++k) {
        float h = fmaxf(x[k] * in + bias[k], 0.0f);
        s += h * W[k * 2 + c];
    }
    Y[idx] = s * onorm[row];
}

// ---------------------------------------------------------------------------
// SpMM scatter: agg[dst[e], f] += Y[src[e], f]  (one thread per (e,f))
// D consecutive lanes share an edge -> coalesced gather + coalesced atomics
// ---------------------------------------------------------------------------
template <int D>
__global__ void spmm_scatter(const float* __restrict__ Y, const int* __restrict__ src,
                             const int* __restrict__ dst, float* __restrict__ agg,
                             long long total) {
    long long idx = (long long)blockIdx.x * blockDim.x + threadIdx.x;
    if (idx >= total) return;
    int e = (int)(idx / D);
    int f = (int)(idx - (long long)e * D);
    int s = src[e];
    int d = dst[e];
    atomicAdd(&agg[(size_t)d * D + f], Y[(size_t)s * D + f]);
}

// ---------------------------------------------------------------------------
// Final epilogue (layer 5, no ReLU): out = agg * in_norm + b
// ---------------------------------------------------------------------------
__global__ void epilogue_final(const float* __restrict__ agg, const float* __restrict__ inorm,
                               const float* __restrict__ b, float* __restrict__ out, int n) {
    int idx = blockIdx.x * blockDim.x + threadIdx.x;
    if (idx >= n * 2) return;
    int row = idx >> 1;
    int f = idx & 1;
    out[idx] = agg[idx] * inorm[row] + b[f];
}

// ---------------------------------------------------------------------------
// Host launcher
// ---------------------------------------------------------------------------
static inline int cdiv(long long a, int b) { return (int)((a + b - 1) / b); }

extern "C" void kernel_launch(void* const* d_in, const int* in_sizes, int n_in,
                              void* d_out, int out_size, void* d_ws, size_t ws_size,
                              hipStream_t stream) {
    const float* in_feat = (const float*)d_in[0];
    const int*   src     = (const int*)d_in[1];
    const int*   dst     = (const int*)d_in[2];
    const float* Wt[5]   = {(const float*)d_in[3], (const float*)d_in[5], (const float*)d_in[7],
                            (const float*)d_in[9], (const float*)d_in[11]};
    const float* bt[5]   = {(const float*)d_in[4], (const float*)d_in[6], (const float*)d_in[8],
                            (const float*)d_in[10], (const float*)d_in[12]};
    float* out = (float*)d_out;

    const int n = in_sizes[0] / 128;   // N nodes
    const int E = in_sizes[1];         // edges

    // Workspace carve-up (floats): norms (2N) + Y buffer + agg buffer (~52 MB)
    float* ws    = (float*)d_ws;
    float* onorm = ws;
    float* inorm = onorm + n;
    float* bufA  = inorm + n;                 // GEMM output (Y)
    float* bufC  = bufA + (size_t)n * 64;     // scatter accumulator (agg)

    // ---- degree norms --------------------------------------------------
    zero_f32<<<cdiv(2LL * n, 256), 256, 0, stream>>>(onorm, 2LL * n);
    degree_kernel<<<cdiv(E, 256), 256, 0, stream>>>(src, dst, onorm, inorm, E);
    norm_kernel<<<cdiv(n, 256), 256, 0, stream>>>(onorm, inorm, n);

    const int gblocks = (n + 31) / 32;
    const long long nd = (long long)n * 64;
    const long long tot64 = (long long)E * 64;

    // ---- layer 1: 128 -> 64 (no fused epilogue) ------------------------
    gemm_wmma<128, false><<<gblocks, 256, 0, stream>>>(in_feat, Wt[0], onorm, inorm, bt[0], bufA, n);
    zero_f32<<<cdiv(nd, 256), 256, 0, stream>>>(bufC, nd);
    spmm_scatter<64><<<cdiv(tot64, 256), 256, 0, stream>>>(bufA, src, dst, bufC, tot64);

    // ---- layers 2..4: 64 -> 64, fusing previous epilogue into staging --
    for (int layer = 1; layer < 4; ++layer) {
        gemm_wmma<64, true><<<gblocks, 256, 0, stream>>>(bufC, Wt[layer], onorm, inorm,
                                                         bt[layer - 1], bufA, n);
        zero_f32<<<cdiv(nd, 256), 256, 0, stream>>>(bufC, nd);
        spmm_scatter<64><<<cdiv(tot64, 256), 256, 0, stream>>>(bufA, src, dst, bufC, tot64);
    }

    // ---- layer 5: 64 -> 2, fusing layer-4 epilogue ---------------------
    const long long tot2 = (long long)E * 2;
    gemm_small<<<cdiv(2LL * n, 256), 256, 0, stream>>>(bufC, Wt[4], onorm, inorm, bt[3], bufA, n);
    zero_f32<<<cdiv(2LL * n, 256), 256, 0, stream>>>(bufC, 2LL * n);
    spmm_scatter<2><<<cdiv(tot2, 256), 256, 0, stream>>>(bufA, src, dst, bufC, tot2);
    epilogue_final<<<cdiv(2LL * n, 256), 256, 0, stream>>>(bufC, inorm, bt[4], out, n);
}